// WaveConv1d_65927747994061
// MI455X (gfx1250) — compile-verified
//
#include <hip/hip_runtime.h>
#include <hip/hip_bf16.h>

// ---------------------------------------------------------------------------
// WaveConv1d (8-level db6 DWT -> channel mixing via WMMA -> inverse DWT)
// MI455X / gfx1250, wave32. fp32 end-to-end (matches reference precision).
// Channel-mixing einsum uses V_WMMA_F32_16X16X4_F32.
// ---------------------------------------------------------------------------

typedef __attribute__((ext_vector_type(2))) float v2f;
typedef __attribute__((ext_vector_type(8))) float v8f;

__device__ constexpr float DECLO[12] = {
    -0.00107730108499558f,  0.004777257511010651f, 0.0005538422009938016f,
    -0.031582039318031156f, 0.02752286553001629f,  0.09750160558707936f,
    -0.12976686756709563f, -0.22626469396516913f,  0.3152503517092432f,
     0.7511339080215775f,   0.4946238903983854f,   0.11154074335008017f};

// analysis taps (conv with DEC_*[::-1], i.e. tap t multiplies xp[2j+t])
__device__ __forceinline__ constexpr float alo_tap(int t) { return DECLO[11 - t]; }
__device__ __forceinline__ constexpr float ahi_tap(int t) {
    return (t & 1) ? -DECLO[t] : DECLO[t];   // DEC_HI[11-t] = (-1)^t * DEC_LO[t]
}
// synthesis taps (transposed conv): tap t multiplies padded-dilated[j+t]
__device__ __forceinline__ constexpr float slo_tap(int t) { return DECLO[t]; }          // REC_LO[11-t]
__device__ __forceinline__ constexpr float shi_tap(int t) {
    return (t & 1) ? DECLO[11 - t] : -DECLO[11 - t];  // REC_HI[11-t] = -(-1)^t DEC_LO[11-t]
}

#define AFB_TILE 256
#define SFB_TILE 256

// ---------------------------------------------------------------------------
// Analysis filter bank: one level of DWT. Depthwise, stride 2, symmetric pad
// (padL = 10 at every level for this config). LDS-staged: 2*TILE+10 inputs
// feed TILE (lo,hi) output pairs.
// ---------------------------------------------------------------------------
__global__ void afb_kernel(const float* __restrict__ in, int Nin,
                           float* __restrict__ lo, float* __restrict__ hi, int Nout) {
    __shared__ float s[2 * AFB_TILE + 12];
    const int bc = blockIdx.y;
    const int j0 = blockIdx.x * AFB_TILE;
    const int j1 = (j0 + AFB_TILE < Nout) ? (j0 + AFB_TILE) : Nout;
    const int span = 2 * (j1 - j0) + 10;
    const float* __restrict__ row = in + (size_t)bc * Nin;

    for (int idx = threadIdx.x; idx < span; idx += 256) {
        int src = 2 * j0 + idx - 10;                  // padded coord - padL
        if (src < 0) src = -src - 1;                  // symmetric reflect (left)
        if (src >= Nin) src = 2 * Nin - 1 - src;      // symmetric reflect (right)
        s[idx] = row[src];
    }
    __syncthreads();

    const int j = j0 + threadIdx.x;
    if (j < Nout) {
        const int base = 2 * threadIdx.x;
        float accL = 0.f, accH = 0.f;
#pragma unroll
        for (int t = 0; t < 12; ++t) {
            const float x = s[base + t];
            accL = fmaf(x, alo_tap(t), accL);
            accH = fmaf(x, ahi_tap(t), accH);
        }
        lo[(size_t)bc * Nout + j] = accL;
        hi[(size_t)bc * Nout + j] = accH;
    }
}

// ---------------------------------------------------------------------------
// Synthesis filter bank: out[j] = sum_{t:(j+t) odd} lo[(j+t-1)/2]*slo(t)
//                                + hi[(j+t-1)/2]*shi(t), Nout = 2n-10.
// lo may be one element longer than n (the reference's trim) -> lo_stride.
// ---------------------------------------------------------------------------
__global__ void sfb_kernel(const float* __restrict__ lo, int lo_stride,
                           const float* __restrict__ hi, int n,
                           float* __restrict__ out, int Nout) {
    __shared__ float slo[SFB_TILE / 2 + 8];
    __shared__ float shi[SFB_TILE / 2 + 8];
    const int bc = blockIdx.y;
    const int j0 = blockIdx.x * SFB_TILE;     // multiple of 256 (even)
    const int i0 = j0 >> 1;
    const int SPAN = SFB_TILE / 2 + 6;        // max k used = TILE/2 + 4

    const float* __restrict__ lrow = lo + (size_t)bc * lo_stride;
    const float* __restrict__ hrow = hi + (size_t)bc * n;
    for (int k = threadIdx.x; k < SPAN; k += 256) {
        const int i = i0 + k;
        const bool ok = (i < n);
        slo[k] = ok ? lrow[i] : 0.f;
        shi[k] = ok ? hrow[i] : 0.f;
    }
    __syncthreads();

    const int j = j0 + threadIdx.x;
    if (j < Nout) {
        float acc = 0.f;
#pragma unroll
        for (int t = 0; t < 12; ++t) {
            if (((j + t) & 1) == 1) {
                const int k = ((j + t - 1) >> 1) - i0;
                acc = fmaf(slo[k], slo_tap(t), acc);
                acc = fmaf(shi[k], shi_tap(t), acc);
            }
        }
        out[(size_t)bc * Nout + j] = acc;
    }
}

// ---------------------------------------------------------------------------
// Channel mixing: Out[b,o,m] = sum_i A[b,i,m] * W[i,o,m], C = 64, M = 42.
// Per x-position m: 64x64x64 fp32 GEMM via V_WMMA_F32_16X16X4_F32.
// One wave = one 16x16 tile, K = 64 as 16 chained k=4 WMMAs.
// grid = (2 tile-groups, 42 positions, 2 tensors), block = 256 (8 waves).
// ---------------------------------------------------------------------------
__global__ void mix_wmma_kernel(const float* __restrict__ A0, const float* __restrict__ W0,
                                float* __restrict__ O0,
                                const float* __restrict__ A1, const float* __restrict__ W1,
                                float* __restrict__ O1) {
    constexpr int M = 42, C = 64;
    const float* __restrict__ A = (blockIdx.z == 0) ? A0 : A1;
    const float* __restrict__ W = (blockIdx.z == 0) ? W0 : W1;
    float* __restrict__ O = (blockIdx.z == 0) ? O0 : O1;

    const int m = blockIdx.y;
    const int wave = threadIdx.x >> 5;
    const int lane = threadIdx.x & 31;
    const int tile = blockIdx.x * 8 + wave;   // 0..15
    const int bt = tile >> 2;                 // batch-row tile
    const int ot = tile & 3;                  // out-channel tile
    const int half = lane >> 4;               // 0: K+{0,1} | 1: K+{2,3}
    const int r = lane & 15;

    const int brow = bt * 16 + r;             // A-matrix row (batch)
    const int ocol = ot * 16 + r;             // B-matrix col (out channel)

    v8f c = {};
#pragma unroll
    for (int kc = 0; kc < 16; ++kc) {
        const int ka = kc * 4 + 2 * half;     // input channel pair base
        v2f a, b;
        // A 16x4 fragment: lanes 0-15 hold K=ka(+1) = chunk K0/K1, lanes 16-31 K2/K3
        a.x = A[(size_t)brow * C * M + (size_t)ka * M + m];
        a.y = A[(size_t)brow * C * M + (size_t)(ka + 1) * M + m];
        // B 4x16 fragment, mirrored row split across lane halves
        b.x = W[(size_t)ka * C * M + (size_t)ocol * M + m];
        b.y = W[(size_t)(ka + 1) * C * M + (size_t)ocol * M + m];
        c = __builtin_amdgcn_wmma_f32_16x16x4_f32(false, a, false, b, (short)0, c,
                                                  false, false);
    }

    const float* cf = (const float*)&c;
#pragma unroll
    for (int v = 0; v < 8; ++v) {
        const int brow_out = bt * 16 + v + half * 8;  // C/D: vgpr v -> M=v / v+8
        O[(size_t)brow_out * C * M + (size_t)(ot * 16 + r) * M + m] = cf[v];
    }
}

// ---------------------------------------------------------------------------
// Host-side orchestration
// ---------------------------------------------------------------------------
extern "C" void kernel_launch(void* const* d_in, const int* in_sizes, int n_in,
                              void* d_out, int out_size, void* d_ws, size_t ws_size,
                              hipStream_t stream) {
    (void)in_sizes; (void)n_in; (void)out_size; (void)ws_size;

    const float* x  = (const float*)d_in[0];
    const float* w1 = (const float*)d_in[1];
    const float* w2 = (const float*)d_in[2];
    float* out = (float*)d_out;
    float* ws = (float*)d_ws;

    const int BC = 64 * 64;
    static const int nsz[9] = {8192, 4101, 2056, 1033, 522, 266, 138, 74, 42};

    // workspace layout (floats): 8 hi buffers, 2 ping-pong lo/recon, 2 mixed
    size_t off = 0;
    float* hbuf[8];
    for (int k = 0; k < 8; ++k) { hbuf[k] = ws + off; off += (size_t)nsz[k + 1] * BC; }
    float* loA = ws + off; off += (size_t)4101 * BC;
    float* loB = ws + off; off += (size_t)4101 * BC;
    float* ylm = ws + off; off += (size_t)42 * BC;
    float* him = ws + off; off += (size_t)42 * BC;
    // total ~271 MB

    const dim3 blk(256);

    // ---- forward DWT: 8 levels, lo ping-pongs A/B, hi -> dedicated buffers
    const float* cur = x;
    float* ping = loA;
    float* pong = loB;
    for (int k = 0; k < 8; ++k) {
        const int Nin = nsz[k], Nout = nsz[k + 1];
        dim3 grd((Nout + AFB_TILE - 1) / AFB_TILE, BC);
        afb_kernel<<<grd, blk, 0, stream>>>(cur, Nin, ping, hbuf[k], Nout);
        cur = ping;
        float* t = ping; ping = pong; pong = t;
    }
    const float* yl = cur;  // = loB after 8 levels (42 per row)

    // ---- channel mixing (WMMA): yl x w1 -> ylm ; hi8 x w2 -> him
    mix_wmma_kernel<<<dim3(2, 42, 2), blk, 0, stream>>>(yl, w1, ylm, hbuf[7], w2, him);

    // ---- inverse DWT: reconstruct, ping-pong loA/loB, final level -> d_out
    // lengths: 42->74->138->266->522->1034->(trim 1033)->2056->4102->(trim 4101)->8192
    sfb_kernel<<<dim3((74   + 255) / 256, BC), blk, 0, stream>>>(ylm,  42,  him,     42,  loA,   74);
    sfb_kernel<<<dim3((138  + 255) / 256, BC), blk, 0, stream>>>(loA,  74,  hbuf[6], 74,  loB,  138);
    sfb_kernel<<<dim3((266  + 255) / 256, BC), blk, 0, stream>>>(loB, 138,  hbuf[5], 138, loA,  266);
    sfb_kernel<<<dim3((522  + 255) / 256, BC), blk, 0, stream>>>(loA, 266,  hbuf[4], 266, loB,  522);
    sfb_kernel<<<dim3((1034 + 255) / 256, BC), blk, 0, stream>>>(loB, 522,  hbuf[3], 522, loA, 1034);
    sfb_kernel<<<dim3((2056 + 255) / 256, BC), blk, 0, stream>>>(loA, 1034, hbuf[2], 1033, loB, 2056);
    sfb_kernel<<<dim3((4102 + 255) / 256, BC), blk, 0, stream>>>(loB, 2056, hbuf[1], 2056, loA, 4102);
    sfb_kernel<<<dim3((8192 + 255) / 256, BC), blk, 0, stream>>>(loA, 4102, hbuf[0], 4101, out, 8192);
}